// PointCloudLoss_57243324121542
// MI455X (gfx1250) — compile-verified
//
#include <hip/hip_runtime.h>
#include <math.h>

typedef __attribute__((ext_vector_type(2))) float v2f;
typedef __attribute__((ext_vector_type(8))) float v8f;

#define NPTS    16384          // K*P points per (cloud,batch)
#define NTILES  1024           // NPTS / 16
#define NB      4              // batches
#define CSTRIDE 16392          // 8*2049 floats between coordinate planes
#define PSTRIDE 2049

// workspace layout (in floats):
//   [0]                       chamfer accumulator
//   [ASQ_OFF  ..)             |p|^2 per point  [2 clouds][4 batches][16384]
//   [AFMT_OFF ..)             WMMA A-format    [2][4][1024 tiles][64]
//   [BFMT_OFF ..)             WMMA B-format    [2][4][1024 tiles][64]
#define ASQ_OFF  16
#define AFMT_OFF (ASQ_OFF + 2 * NB * NPTS)
#define BFMT_OFF (AFMT_OFF + 2 * NB * NTILES * 64)

__global__ void pcl_init(float* ws) {
  if (threadIdx.x == 0 && blockIdx.x == 0) ws[0] = 0.f;
}

// Pack both clouds into WMMA-native layouts.
// A (16x4 f32): lane<16 -> (K0,K1) = (-2x,-2y); lane>=16 -> (K2,K3) = (-2z, 1)
// B (4x16 f32): lane<16 -> (K0,K1) = ( x,  y); lane>=16 -> (K2,K3) = ( z, |b|^2)
__global__ void pcl_pack(const float* __restrict__ pred,
                         const float* __restrict__ tgt,
                         float* __restrict__ ws) {
  int tid = blockIdx.x * blockDim.x + threadIdx.x;   // 0 .. 131071
  int cl = tid >> 16;                                // 0=pred, 1=tgt
  int b  = (tid >> 14) & 3;
  int n  = tid & (NPTS - 1);
  int k = n >> 11, p = n & 2047;
  const float* src = cl ? tgt : pred;
  size_t base = (size_t)(b * 3) * CSTRIDE + (size_t)k * PSTRIDE + p;
  float x = src[base];
  float y = src[base + CSTRIDE];
  float z = src[base + 2 * CSTRIDE];
  float sq = x * x + y * y + z * z;
  int clb = cl * NB + b;
  ws[ASQ_OFF + clb * NPTS + n] = sq;
  int tile = n >> 4, col = n & 15;
  float* Af = ws + AFMT_OFF + (size_t)(clb * NTILES + tile) * 64;
  float* Bf = ws + BFMT_OFF + (size_t)(clb * NTILES + tile) * 64;
  v2f a_lo = { -2.f * x, -2.f * y };
  v2f a_hi = { -2.f * z, 1.0f };
  v2f b_lo = { x, y };
  v2f b_hi = { z, sq };
  *(v2f*)(Af + 2 * col)        = a_lo;
  *(v2f*)(Af + 2 * (16 + col)) = a_hi;
  *(v2f*)(Bf + 2 * col)        = b_lo;
  *(v2f*)(Bf + 2 * (16 + col)) = b_hi;
}

#define ATPW 4   // query tiles per wave (amortizes B-tile loads 4x)

// One wave: 4 query tiles (64 points) of one (batch,direction); streams all
// 1024 database tiles. D = A*B gives -2 a.b + |b|^2 per 16x16 tile; running
// elementwise min; |a|^2 added after the cross-lane min, then sqrt.
__global__ void __launch_bounds__(256) pcl_chamfer(float* __restrict__ ws) {
  const float* asq  = ws + ASQ_OFF;
  const float* Afmt = ws + AFMT_OFF;
  const float* Bfmt = ws + BFMT_OFF;
  int lane = threadIdx.x & 31;
  int gw = blockIdx.x * (blockDim.x >> 5) + (threadIdx.x >> 5);  // 0..2047
  int dir   = gw >> 10;            // 0: pred->tgt, 1: tgt->pred
  int rem   = gw & 1023;
  int batch = rem >> 8;
  int quad  = rem & 255;
  int clA = dir, clB = 1 - dir;
  int aTile0 = quad * ATPW;
  const float* Af   = Afmt + (size_t)((clA * NB + batch) * NTILES) * 64;
  const float* Bf   = Bfmt + (size_t)((clB * NB + batch) * NTILES) * 64;
  const float* asqA = asq + (clA * NB + batch) * NPTS;

  v2f A[ATPW];
  v8f acc[ATPW];
#pragma unroll
  for (int j = 0; j < ATPW; ++j) {
    A[j] = *(const v2f*)(Af + (size_t)(aTile0 + j) * 64 + 2 * lane);
#pragma unroll
    for (int i = 0; i < 8; ++i) acc[j][i] = 3.0e38f;
  }
  v8f Cz = {0.f, 0.f, 0.f, 0.f, 0.f, 0.f, 0.f, 0.f};

  for (int bt = 0; bt < NTILES; bt += 2) {
    v2f B0 = *(const v2f*)(Bf + (size_t)bt * 64 + 2 * lane);
    v2f B1 = *(const v2f*)(Bf + (size_t)(bt + 1) * 64 + 2 * lane);
#pragma unroll
    for (int j = 0; j < ATPW; ++j) {
      v8f D0 = __builtin_amdgcn_wmma_f32_16x16x4_f32(false, A[j], false, B0,
                                                     (short)0, Cz, false, false);
      v8f D1 = __builtin_amdgcn_wmma_f32_16x16x4_f32(false, A[j], false, B1,
                                                     (short)0, Cz, false, false);
#pragma unroll
      for (int i = 0; i < 8; ++i)
        acc[j][i] = fminf(fminf(D0[i], D1[i]), acc[j][i]);   // min3 pattern
    }
  }

  // cross-lane min within each 16-lane half (C/D layout: VGPR r holds rows r
  // in lanes 0-15 and r+8 in lanes 16-31), then + |a|^2, clamp, sqrt, sum.
  int hbase = (lane & 16) ? 8 : 0;
  float s = 0.f;
#pragma unroll
  for (int j = 0; j < ATPW; ++j) {
#pragma unroll
    for (int r = 0; r < 8; ++r) {
      float v = acc[j][r];
      v = fminf(v, __shfl_xor(v, 1, 32));
      v = fminf(v, __shfl_xor(v, 2, 32));
      v = fminf(v, __shfl_xor(v, 4, 32));
      v = fminf(v, __shfl_xor(v, 8, 32));
      float aq = asqA[(aTile0 + j) * 16 + hbase + r];
      s += sqrtf(fmaxf(v + aq, 0.f));
    }
  }
  s += __shfl_xor(s, 16, 32);
  if (lane == 0) atomicAdd(ws, s);
}

// One wave: 32-element BCE-with-logits detection loss + combine with chamfer.
__global__ void pcl_finalize(const float* __restrict__ pred,
                             const float* __restrict__ tgt,
                             const float* __restrict__ ws,
                             float* __restrict__ out) {
  int lane = threadIdx.x;          // 0..31, one wave32
  int b = lane >> 3, k = lane & 7;
  size_t idx = (size_t)(b * 3) * CSTRIDE + (size_t)k * PSTRIDE + 2048;
  float z = pred[idx];
  float t = tgt[idx] > 0.5f ? 1.f : 0.f;
  float e = fmaxf(z, 0.f) - z * t + log1pf(expf(-fabsf(z)));
  e += __shfl_xor(e, 1, 32);
  e += __shfl_xor(e, 2, 32);
  e += __shfl_xor(e, 4, 32);
  e += __shfl_xor(e, 8, 32);
  e += __shfl_xor(e, 16, 32);
  if (lane == 0)
    out[0] = e * (1.f / 32.f) + ws[0] * (1.f / (2.f * NB * NPTS));
}

extern "C" void kernel_launch(void* const* d_in, const int* in_sizes, int n_in,
                              void* d_out, int out_size, void* d_ws, size_t ws_size,
                              hipStream_t stream) {
  (void)in_sizes; (void)n_in; (void)out_size; (void)ws_size;
  const float* pred = (const float*)d_in[0];
  const float* tgt  = (const float*)d_in[1];
  float* ws  = (float*)d_ws;
  float* out = (float*)d_out;

  hipLaunchKernelGGL(pcl_init, dim3(1), dim3(1), 0, stream, ws);
  hipLaunchKernelGGL(pcl_pack, dim3(512), dim3(256), 0, stream, pred, tgt, ws);
  hipLaunchKernelGGL(pcl_chamfer, dim3(256), dim3(256), 0, stream, ws);
  hipLaunchKernelGGL(pcl_finalize, dim3(1), dim3(32), 0, stream, pred, tgt, ws, out);
}